// NCO_59081570124226
// MI455X (gfx1250) — compile-verified
//
#include <hip/hip_runtime.h>
#include <hip/hip_bf16.h>
#include <math.h>

// ---------------------------------------------------------------------------
// NCO pipeline for MI455X (gfx1250, wave32).
//   Stage 1: stds + correlation matrix
//   Stage 2: k-means (K=5) on corr rows; distances via fp32 WMMA GEMM
//   Stage 3: per-cluster min-var weights via masked-SPD conjugate gradient;
//            each CG step = one Sigma @ (1024x5) GEMM via V_WMMA_F32_16X16X4_F32
//   Stage 4: inter-cluster covariance (WMMA GEMM) + 5x5 inverse + blend
//
// B panels in LDS carry a permanently-zero 6th column (BST=6) so WMMA lanes
// covering padded N columns (5..15) read zeros directly (no mask multiply on
// the ds_load -> v_wmma critical path). Each wave computes TWO 16-row tiles
// in one fused K-loop so every B fragment loaded from LDS feeds two WMMAs.
// ---------------------------------------------------------------------------

#define NS 8          // samples
#define NA 1024       // assets
#define NK 5          // clusters
#define BST 6         // B-panel LDS stride (col NK is the zero column)
#define KM_ITERS 30
#define CG_ITERS 96

typedef __attribute__((ext_vector_type(2))) float v2f;
typedef __attribute__((ext_vector_type(8))) float v8f;

// ---------------------------------------------------------------------------
// Wave-cooperative fused 2-tile fp32 WMMA GEMM: rows [m0, m0+32) of C = A @ B.
//   A : global row-major [NA][NA]
//   B : LDS [NA][BST], columns 0..NK-1 valid, column NK always zero
//   C : LDS [NA][NK]
//
// V_WMMA_F32_16X16X4_F32 operand layout (wave32):
//   A 16x4 : lanes 0-15 rows M=0..15, vgpr{0,1} = K{0,1}; lanes 16-31 K{2,3}
//   B 4x16 : lanes 0-15 cols N=0..15, vgpr{0,1} = K{0,1}; lanes 16-31 K{2,3}
//   C 16x16: vgpr v -> row v (lanes 0-15) / row 8+v (lanes 16-31), col = lane%16
// ---------------------------------------------------------------------------
__device__ inline void gemm_tile2_f32(const float* __restrict__ Abase,
                                      int m0,
                                      const float* __restrict__ Blds,
                                      float* __restrict__ Clds)
{
    const int lane = threadIdx.x & 31;
    const int half = lane >> 4;                 // K-group select
    const int l16  = lane & 15;                 // row (A) / col (B,C)
    const float* arow0 = Abase + (size_t)(m0 + l16) * NA;
    const float* arow1 = arow0 + (size_t)16 * NA;
    const int bcol = (l16 < NK) ? l16 : NK;     // padded lanes hit zero column

    v8f acc0 = {}, acc1 = {};
    for (int k = 0; k < NA; k += 4) {
        v2f a0, a1, b;
        b[0]  = Blds[(k + 2 * half + 0) * BST + bcol];
        b[1]  = Blds[(k + 2 * half + 1) * BST + bcol];
        a0[0] = arow0[k + 2 * half + 0];
        a0[1] = arow0[k + 2 * half + 1];
        a1[0] = arow1[k + 2 * half + 0];
        a1[1] = arow1[k + 2 * half + 1];
        acc0 = __builtin_amdgcn_wmma_f32_16x16x4_f32(
            false, a0, false, b, (short)0, acc0, false, false);
        acc1 = __builtin_amdgcn_wmma_f32_16x16x4_f32(
            false, a1, false, b, (short)0, acc1, false, false);
    }
    if (l16 < NK) {
        #pragma unroll
        for (int v = 0; v < 8; ++v) {
            Clds[(m0 + half * 8 + v) * NK + l16]      = acc0[v];
            Clds[(m0 + 16 + half * 8 + v) * NK + l16] = acc1[v];
        }
    }
}

// ---------------------------------------------------------------------------
// Stage 1a: stds[s][a] = sqrt(cov[s][a][a])
// ---------------------------------------------------------------------------
__global__ __launch_bounds__(NA) void nco_stds_kernel(const float* __restrict__ cov,
                                                      float* __restrict__ stds)
{
    const int s = blockIdx.x, a = threadIdx.x;
    stds[s * NA + a] = sqrtf(cov[(size_t)s * NA * NA + (size_t)a * NA + a]);
}

// ---------------------------------------------------------------------------
// Stage 1b: corr[s][i][j] = cov[s][i][j] / (std_i * std_j)
// ---------------------------------------------------------------------------
__global__ __launch_bounds__(NA) void nco_corr_kernel(const float* __restrict__ cov,
                                                      const float* __restrict__ stds,
                                                      float* __restrict__ corr)
{
    const int i = blockIdx.x, s = blockIdx.y, j = threadIdx.x;
    const size_t off = (size_t)s * NA * NA + (size_t)i * NA + j;
    corr[off] = cov[off] / (stds[s * NA + i] * stds[s * NA + j]);
}

// ---------------------------------------------------------------------------
// Stage 2: k-means on rows of corr. One workgroup (32 waves) per sample.
// Distances: d = |x|^2 + |c|^2 - 2 x.c ; the x@C^T GEMM runs on WMMA.
// GS doubles as the distance Gram (G) and the scatter-mean accumulator
// (csum): thread t reads only its own G row, zeroes it, barrier, reuse.
// ---------------------------------------------------------------------------
__global__ __launch_bounds__(NA) void nco_kmeans_kernel(const float* __restrict__ corr,
                                                        int* __restrict__ ixs_g)
{
    __shared__ float ct[NA * BST];    // centers transposed [dim][k], col NK = 0
    __shared__ float GS[NA * NK];     // Gram, then per-cluster coordinate sums
    __shared__ float cn[NK];          // center squared norms
    __shared__ float cnt[NK];         // cluster counts
    __shared__ float s_pot;
    __shared__ int   s_stop;

    const int s = blockIdx.x, t = threadIdx.x;
    const float* X = corr + (size_t)s * NA * NA;

    // deterministic spread init (JAX PRNG choice not reproducible on-device)
    const int init_idx[NK] = {71, 263, 467, 683, 911};
    #pragma unroll
    for (int k = 0; k < NK; ++k)
        ct[t * BST + k] = X[(size_t)init_idx[k] * NA + t];
    ct[t * BST + NK] = 0.0f;          // permanent zero column

    // per-point squared norm (thread t owns point t)
    float xn = 0.0f;
    for (int j = 0; j < NA; ++j) { float v = X[(size_t)t * NA + j]; xn += v * v; }

    float prev_pot = __builtin_inff();
    int my_ix = 0;
    if (t == 0) s_stop = 0;
    __syncthreads();

    for (int it = 0; it < KM_ITERS; ++it) {
        if (t < NK) { cn[t] = 0.0f; cnt[t] = 0.0f; }
        if (t == 0) s_pot = 0.0f;
        __syncthreads();

        #pragma unroll
        for (int k = 0; k < NK; ++k) { float c = ct[t * BST + k]; atomicAdd(&cn[k], c * c); }
        __syncthreads();

        // G = X @ centers^T   (WMMA, fused 2-tile GEMM per wave)
        gemm_tile2_f32(X, (t >> 5) * 32, ct, GS);
        __syncthreads();

        // assignment + potential (pull own Gram row into registers)
        float g[NK];
        #pragma unroll
        for (int k = 0; k < NK; ++k) g[k] = GS[t * NK + k];
        float best = __builtin_inff(); int bix = 0;
        #pragma unroll
        for (int k = 0; k < NK; ++k) {
            float d = xn + cn[k] - 2.0f * g[k];
            d = d < 0.0f ? 0.0f : d;
            if (d < best) { best = d; bix = k; }
        }
        my_ix = bix;
        atomicAdd(&s_pot, best);
        atomicAdd(&cnt[bix], 1.0f);
        // repurpose GS as csum: zero own row (own-row read already done)
        #pragma unroll
        for (int k = 0; k < NK; ++k) GS[t * NK + k] = 0.0f;
        __syncthreads();

        for (int j = 0; j < NA; ++j)
            atomicAdd(&GS[j * NK + bix], X[(size_t)t * NA + j]);
        __syncthreads();

        // centroid update (thread t owns dim t)
        #pragma unroll
        for (int k = 0; k < NK; ++k) {
            float c = cnt[k];
            if (c > 0.0f) ct[t * BST + k] = GS[t * NK + k] / c;
        }
        if (t == 0) {
            float pot = s_pot;
            if (it > 0 && fabsf(pot - prev_pot) < 1e-5f) s_stop = 1;
            prev_pot = pot;
        }
        __syncthreads();
        if (s_stop) break;
    }
    ixs_g[s * NA + t] = my_ix;
}

// ---------------------------------------------------------------------------
// Stage 3: batched CG for the 5 masked SPD systems of one sample.
// M_k v = m_k .* (Sigma (m_k .* v)) + (1-m_k) .* v  -> one shared-Sigma GEMM
// per CG step for all 5 clusters. Produces intra-cluster weights.
// ---------------------------------------------------------------------------
__global__ __launch_bounds__(NA) void nco_cg_kernel(const float* __restrict__ cov,
                                                    const int* __restrict__ ixs_g,
                                                    float* __restrict__ Wbuf)
{
    __shared__ float U[NA * BST];     // masked search directions (B panel)
    __shared__ float Y[NA * NK];      // Sigma @ U
    __shared__ float rrs[NK], paps[NK], rrns[NK], dns[NK];

    const int s = blockIdx.x, t = threadIdx.x;
    const float* Sg = cov + (size_t)s * NA * NA;
    const int ixt = ixs_g[s * NA + t];

    float x[NK], r[NK], p[NK];
    #pragma unroll
    for (int k = 0; k < NK; ++k) {
        float m = (ixt == k) ? 1.0f : 0.0f;
        x[k] = 0.0f; r[k] = m; p[k] = m;         // b = mask vector
    }
    U[t * BST + NK] = 0.0f;                      // permanent zero column
    if (t < NK) rrs[t] = 0.0f;
    __syncthreads();
    #pragma unroll
    for (int k = 0; k < NK; ++k) atomicAdd(&rrs[k], r[k] * r[k]);
    __syncthreads();

    for (int iter = 0; iter < CG_ITERS; ++iter) {
        #pragma unroll
        for (int k = 0; k < NK; ++k) U[t * BST + k] = (ixt == k) ? p[k] : 0.0f;
        if (t < NK) { paps[t] = 0.0f; rrns[t] = 0.0f; }
        __syncthreads();

        gemm_tile2_f32(Sg, (t >> 5) * 32, U, Y);
        __syncthreads();

        float Ap[NK];
        #pragma unroll
        for (int k = 0; k < NK; ++k) Ap[k] = (ixt == k) ? Y[t * NK + k] : p[k];
        #pragma unroll
        for (int k = 0; k < NK; ++k) atomicAdd(&paps[k], p[k] * Ap[k]);
        __syncthreads();

        #pragma unroll
        for (int k = 0; k < NK; ++k) {
            float pap = paps[k];
            float alpha = (pap != 0.0f) ? rrs[k] / pap : 0.0f;
            x[k] += alpha * p[k];
            r[k] -= alpha * Ap[k];
        }
        #pragma unroll
        for (int k = 0; k < NK; ++k) atomicAdd(&rrns[k], r[k] * r[k]);
        __syncthreads();
        #pragma unroll
        for (int k = 0; k < NK; ++k) {
            float rr = rrs[k];
            float beta = (rr != 0.0f) ? rrns[k] / rr : 0.0f;
            p[k] = r[k] + beta * p[k];
        }
        __syncthreads();
        if (t < NK) rrs[t] = rrns[t];
        __syncthreads();
    }

    // intra-cluster weights: w = m.*x / sum(m.*x)
    if (t < NK) dns[t] = 0.0f;
    __syncthreads();
    #pragma unroll
    for (int k = 0; k < NK; ++k) if (ixt == k) atomicAdd(&dns[k], x[k]);
    __syncthreads();
    #pragma unroll
    for (int k = 0; k < NK; ++k) {
        float w = (ixt == k && dns[k] != 0.0f) ? x[k] / dns[k] : 0.0f;
        Wbuf[((size_t)s * NA + t) * NK + k] = w;
    }
}

// ---------------------------------------------------------------------------
// Stage 4: inter = W^T Sigma W (5x5); min-var over clusters; blend output.
// ---------------------------------------------------------------------------
__global__ __launch_bounds__(NA) void nco_inter_kernel(const float* __restrict__ cov,
                                                       const float* __restrict__ Wbuf,
                                                       float* __restrict__ out)
{
    __shared__ float Wl[NA * BST];    // intra weights (B panel), col NK = 0
    __shared__ float Tl[NA * NK];     // Sigma @ W
    __shared__ float inter[NK * NK];
    __shared__ float winter[NK];

    const int s = blockIdx.x, t = threadIdx.x;
    const float* Sg = cov + (size_t)s * NA * NA;

    #pragma unroll
    for (int k = 0; k < NK; ++k) Wl[t * BST + k] = Wbuf[((size_t)s * NA + t) * NK + k];
    Wl[t * BST + NK] = 0.0f;
    if (t < NK * NK) inter[t] = 0.0f;
    __syncthreads();

    gemm_tile2_f32(Sg, (t >> 5) * 32, Wl, Tl);
    __syncthreads();

    #pragma unroll
    for (int k = 0; k < NK; ++k)
        #pragma unroll
        for (int l = 0; l < NK; ++l)
            atomicAdd(&inter[k * NK + l], Wl[t * BST + k] * Tl[t * NK + l]);
    __syncthreads();

    if (t == 0) {
        // Gauss-Jordan inverse of 5x5 with partial pivoting
        float M[NK][2 * NK];
        for (int i = 0; i < NK; ++i)
            for (int j = 0; j < NK; ++j) {
                M[i][j] = inter[i * NK + j];
                M[i][NK + j] = (i == j) ? 1.0f : 0.0f;
            }
        for (int c = 0; c < NK; ++c) {
            int piv = c; float mx = fabsf(M[c][c]);
            for (int rr2 = c + 1; rr2 < NK; ++rr2) {
                float v = fabsf(M[rr2][c]);
                if (v > mx) { mx = v; piv = rr2; }
            }
            if (piv != c)
                for (int j = 0; j < 2 * NK; ++j) {
                    float tmp = M[c][j]; M[c][j] = M[piv][j]; M[piv][j] = tmp;
                }
            float d = M[c][c];
            float inv = (d != 0.0f) ? 1.0f / d : 0.0f;
            for (int j = 0; j < 2 * NK; ++j) M[c][j] *= inv;
            for (int rr2 = 0; rr2 < NK; ++rr2) {
                if (rr2 == c) continue;
                float f = M[rr2][c];
                for (int j = 0; j < 2 * NK; ++j) M[rr2][j] -= f * M[c][j];
            }
        }
        // w_inter = rowsum(inv) / total
        float rs[NK], tot = 0.0f;
        for (int i = 0; i < NK; ++i) {
            float a = 0.0f;
            for (int j = 0; j < NK; ++j) a += M[i][NK + j];
            rs[i] = a; tot += a;
        }
        float itot = (tot != 0.0f) ? 1.0f / tot : 0.0f;
        for (int i = 0; i < NK; ++i) winter[i] = rs[i] * itot;
    }
    __syncthreads();

    float o = 0.0f;
    #pragma unroll
    for (int k = 0; k < NK; ++k) o += Wl[t * BST + k] * winter[k];
    out[s * NA + t] = o;
}

// ---------------------------------------------------------------------------
extern "C" void kernel_launch(void* const* d_in, const int* in_sizes, int n_in,
                              void* d_out, int out_size, void* d_ws, size_t ws_size,
                              hipStream_t stream)
{
    (void)in_sizes; (void)n_in; (void)out_size; (void)ws_size;
    const float* cov = (const float*)d_in[0];     // (8, 1024, 1024) fp32
    float* out = (float*)d_out;                   // (8, 1024) fp32

    // workspace carve: corr (32MB) | stds (32KB) | ixs (32KB) | W (160KB)
    float* corr = (float*)d_ws;
    float* stds = corr + (size_t)NS * NA * NA;
    int*   ixs  = (int*)(stds + (size_t)NS * NA);
    float* Wbuf = (float*)(ixs + (size_t)NS * NA);

    nco_stds_kernel<<<NS, NA, 0, stream>>>(cov, stds);
    nco_corr_kernel<<<dim3(NA, NS), NA, 0, stream>>>(cov, stds, corr);
    nco_kmeans_kernel<<<NS, NA, 0, stream>>>(corr, ixs);
    nco_cg_kernel<<<NS, NA, 0, stream>>>(cov, ixs, Wbuf);
    nco_inter_kernel<<<NS, NA, 0, stream>>>(cov, Wbuf, out);
}